// TweetRep_26860725469710
// MI455X (gfx1250) — compile-verified
//
#include <hip/hip_runtime.h>

typedef float v2f __attribute__((ext_vector_type(2)));
typedef float v8f __attribute__((ext_vector_type(8)));

#define SEQN  20
#define EMBN  32
#define WAVES 8   // waves (= w-positions) per block

// One wave = one site (b,l,h,w); lane = embedding dim e (EMB==32==wave32).
// Scores b_s = sum_e p[e][s] are computed with exact-f32 V_WMMA_F32_16X16X4_F32
// chains: A = ones(16x4) makes D = column sums of B, accumulated over 8 K-chunks
// covering all 32 lanes' contributions. This is invariant to the (unverified)
// B-operand K-slot mapping, and uses only the C/D layout for
// extraction (row 0 lives in acc[0], lanes 0..15 hold N=0..15).
__global__ __launch_bounds__(256) void tweet_pool_kernel(
    const int* __restrict__ x,
    const float* __restrict__ emb,
    float* __restrict__ out)
{
    __shared__ float sm_p[WAVES * 1024];  // per-wave 32x32 p matrix, xor-swizzled
    __shared__ float sm_b[WAVES * 32];    // per-wave scores
    __shared__ float sm_o[EMBN * 9];      // cross-wave output transpose (pad 9)

    const int tid  = threadIdx.x;
    const int lane = tid & 31;            // embedding dim e
    const int wv   = tid >> 5;            // wave id = w offset within block
    const int g    = blockIdx.x;

    // Wave-uniform site index -> scalar-path index loads.
    const int site = __builtin_amdgcn_readfirstlane(g * WAVES + wv);
    const int* xp  = x + (size_t)site * SEQN;

    // Gather: one 128B cacheline per token, fully coalesced (lane = dim).
    float w[SEQN];
#pragma unroll
    for (int s = 0; s < SEQN; ++s) {
        const int idx = xp[s];
        w[s] = emb[(size_t)idx * EMBN + lane];
    }
    float V = 0.0f;
#pragma unroll
    for (int s = 0; s < SEQN; ++s) V += w[s];

    // p[e][s] = Wt[s][e] * V[e] into LDS; swizzle addr = e*32 + (s^e);
    // zero-pad s = 20..31 so the second WMMA chain reads defined zeros.
    float* pw = sm_p + wv * 1024;
#pragma unroll
    for (int s = 0; s < 32; ++s) {
        const float v = (s < SEQN) ? w[s] * V : 0.0f;
        pw[lane * 32 + (s ^ lane)] = v;
    }

    // b_n (chain1: n=0..15) and b_{16+n} (chain2) via ones-matrix WMMA.
    // B-slot assignment: VGPR0 <- e = 4c+half, VGPR1 <- e = 4c+2+half, which
    // puts the two half-waves on even/odd e -> opposite 32-bank halves.
    const int n    = lane & 15;
    const int half = lane >> 4;
    const v2f ones = {1.0f, 1.0f};
    v8f acc1 = {0.f,0.f,0.f,0.f,0.f,0.f,0.f,0.f};
    v8f acc2 = {0.f,0.f,0.f,0.f,0.f,0.f,0.f,0.f};
#pragma unroll
    for (int c = 0; c < 8; ++c) {
        const int e0 = 4 * c + half;
        const int e1 = 4 * c + 2 + half;
        v2f b1; b1.x = pw[e0 * 32 + (n ^ e0)];
                b1.y = pw[e1 * 32 + (n ^ e1)];
        v2f b2; b2.x = pw[e0 * 32 + ((16 + n) ^ e0)];
                b2.y = pw[e1 * 32 + ((16 + n) ^ e1)];
        acc1 = __builtin_amdgcn_wmma_f32_16x16x4_f32(
                   false, ones, false, b1, (short)0, acc1, false, false);
        acc2 = __builtin_amdgcn_wmma_f32_16x16x4_f32(
                   false, ones, false, b2, (short)0, acc2, false, false);
    }

    // D row M=0 sits in acc[0] for lanes 0..15 (N = lane).
    float* bw = sm_b + wv * 32;
    if (lane < 16) {
        bw[lane]      = acc1[0];
        bw[16 + lane] = acc2[0];
    }

    // Softmax over seq, redundantly per lane (broadcast LDS reads are free).
    float bs[SEQN];
#pragma unroll
    for (int s = 0; s < SEQN; ++s) bs[s] = bw[s];
    float m = bs[0];
#pragma unroll
    for (int s = 1; s < SEQN; ++s) m = fmaxf(m, bs[s]);
    float Z = 0.0f;
#pragma unroll
    for (int s = 0; s < SEQN; ++s) { bs[s] = __expf(bs[s] - m); Z += bs[s]; }
    const float invZ = 1.0f / Z;

    // pooled[e] = sum_s c_s * Wt[s][e]  (pure lane-local FMA chain)
    float pooled = 0.0f;
#pragma unroll
    for (int s = 0; s < SEQN; ++s) pooled = fmaf(w[s], bs[s] * invZ, pooled);

    // Cross-wave transpose so the channel-major store is contiguous:
    // out[b][l*32+e][h][w0+i] = base + e*1024 + i, i = w offset 0..7.
    sm_o[lane * 9 + wv] = pooled;
    __syncthreads();
    const int e = tid >> 3;
    const int i = tid & 7;
    // g = ((b*4+l)*32 + h)*4 + wblock  ->  base = (g>>7)*32768 + (g&127)*8
    const size_t base = (size_t)(g >> 7) * 32768 + (size_t)(g & 127) * 8;
    out[base + (size_t)e * 1024 + i] = sm_o[e * 9 + i];
}

extern "C" void kernel_launch(void* const* d_in, const int* in_sizes, int n_in,
                              void* d_out, int out_size, void* d_ws, size_t ws_size,
                              hipStream_t stream) {
    (void)in_sizes; (void)n_in; (void)d_ws; (void)ws_size; (void)out_size;
    const int*   x   = (const int*)d_in[0];     // (16,4,32,32,20) int32
    const float* emb = (const float*)d_in[1];   // (50001,32) float32
    float*       out = (float*)d_out;           // (16,128,32,32) float32

    // 65536 sites / 8 waves per block = 8192 blocks of 256 threads.
    tweet_pool_kernel<<<8192, 256, 0, stream>>>(x, emb, out);
}